// HalfEdgeConv_63668595196147
// MI455X (gfx1250) — compile-verified
//
#include <hip/hip_runtime.h>

typedef __attribute__((ext_vector_type(2))) float v2f;
typedef __attribute__((ext_vector_type(8))) float v8f;

#define CCH 64          // channels
#define KTOT 128        // concat K
#define TILE_M 16
#define BLOCK 256
#define WAVES_PER_BLOCK 8
#define MAX_BLOCKS 1024

// LDS holds W^T pre-swizzled into exact B-fragment order:
//   frag float index = ((nt*32 + ks)*32 + lane)*2 + e
// where for W element (n, kk):  nt = n>>4, ks = kk>>2,
//   lane = (n&15) | (((kk>>1)&1)<<4), e = kk&1
// This matches the ISA 16x4 B layout: VGPR0 = K 0/2 (lane halves), VGPR1 = K 1/3.

__global__ __launch_bounds__(BLOCK) void heconv_wmma_kernel(
    const float* __restrict__ x,
    const int* __restrict__ next_idx,
    const int* __restrict__ twin_idx,
    const unsigned char* __restrict__ has_twin,
    const float* __restrict__ W,
    const float* __restrict__ bias,
    float* __restrict__ out,
    int n_rows, int n_tiles)
{
    __shared__ float lds_b[4 * 32 * 32 * 2];   // 8192 floats = 32 KB

    const int tid  = threadIdx.x;
    const int lane = tid & 31;
    const int wave = tid >> 5;

    // ---- stage W -> LDS in fragment order (one pass per block) ----
    for (int f = tid; f < CCH * KTOT; f += BLOCK) {
        int n  = f >> 7;          // row of W  (output channel)
        int kk = f & 127;         // col of W  (concat-K)
        int nt = n >> 4;
        int ks = kk >> 2;
        int L  = (n & 15) | (((kk >> 1) & 1) << 4);
        int e  = kk & 1;
        lds_b[((((nt * 32 + ks) * 32 + L) << 1) + e)] = W[f];
    }
    __syncthreads();

    const int lhalf = lane >> 4;      // 0 or 1
    const int l15   = lane & 15;
    const int aoff  = lhalf * 2;      // A fragment K offset inside a 4-wide step

    // bias per output column (identical across the 8 row-VGPRs of each acc tile)
    const float b0 = bias[      l15];
    const float b1 = bias[16 +  l15];
    const float b2 = bias[32 +  l15];
    const float b3 = bias[48 +  l15];

    const int wave_global = blockIdx.x * WAVES_PER_BLOCK + wave;
    const int wave_stride = gridDim.x * WAVES_PER_BLOCK;

#define WMMA_STEP(AV, KS)                                                                 \
    {                                                                                     \
        v2f bf;                                                                           \
        bf = *(const v2f*)(lds_b + (((0 * 32 + (KS)) * 32 + lane) << 1));                 \
        acc0 = __builtin_amdgcn_wmma_f32_16x16x4_f32(false, (AV), false, bf, (short)0,    \
                                                     acc0, false, false);                 \
        bf = *(const v2f*)(lds_b + (((1 * 32 + (KS)) * 32 + lane) << 1));                 \
        acc1 = __builtin_amdgcn_wmma_f32_16x16x4_f32(false, (AV), false, bf, (short)0,    \
                                                     acc1, false, false);                 \
        bf = *(const v2f*)(lds_b + (((2 * 32 + (KS)) * 32 + lane) << 1));                 \
        acc2 = __builtin_amdgcn_wmma_f32_16x16x4_f32(false, (AV), false, bf, (short)0,    \
                                                     acc2, false, false);                 \
        bf = *(const v2f*)(lds_b + (((3 * 32 + (KS)) * 32 + lane) << 1));                 \
        acc3 = __builtin_amdgcn_wmma_f32_16x16x4_f32(false, (AV), false, bf, (short)0,    \
                                                     acc3, false, false);                 \
    }

    for (int t = wave_global; t < n_tiles; t += wave_stride) {
        int r = t * TILE_M + l15;
        int rc = r < n_rows ? r : (n_rows - 1);      // clamp (tail safety)

        const float* pn = x + (long)next_idx[rc] * CCH;
        const float* pt = x + (long)twin_idx[rc] * CCH;
        const float  m  = has_twin[rc] ? 1.0f : 0.0f;

        v8f acc0 = { b0, b0, b0, b0, b0, b0, b0, b0 };
        v8f acc1 = { b1, b1, b1, b1, b1, b1, b1, b1 };
        v8f acc2 = { b2, b2, b2, b2, b2, b2, b2, b2 };
        v8f acc3 = { b3, b3, b3, b3, b3, b3, b3, b3 };

        // K = 0..63 : next-vertex features (gathered row doubles as A fragment)
#pragma unroll
        for (int ks = 0; ks < 16; ++ks) {
            v2f a = *(const v2f*)(pn + ks * 4 + aoff);
            WMMA_STEP(a, ks);
        }
        // K = 64..127 : twin-vertex features, zero-masked per row
#pragma unroll
        for (int ks = 0; ks < 16; ++ks) {
            v2f a = *(const v2f*)(pt + ks * 4 + aoff);
            a = a * m;
            WMMA_STEP(a, ks + 16);
        }

        // epilogue: ReLU + store. D layout: VGPR j -> row (j + 8*lhalf), col = l15 (+16*nt)
        float* obase = out + (long)t * TILE_M * CCH;
#pragma unroll
        for (int j = 0; j < 8; ++j) {
            int row = j + 8 * lhalf;
            if (t * TILE_M + row < n_rows) {
                float* po = obase + (long)row * CCH + l15;
                po[0]  = fmaxf(acc0[j], 0.0f);
                po[16] = fmaxf(acc1[j], 0.0f);
                po[32] = fmaxf(acc2[j], 0.0f);
                po[48] = fmaxf(acc3[j], 0.0f);
            }
        }
    }
#undef WMMA_STEP
}

extern "C" void kernel_launch(void* const* d_in, const int* in_sizes, int n_in,
                              void* d_out, int out_size, void* d_ws, size_t ws_size,
                              hipStream_t stream) {
    const float*         x        = (const float*)d_in[0];
    const int*           next_idx = (const int*)d_in[1];
    const int*           twin_idx = (const int*)d_in[2];
    const unsigned char* has_twin = (const unsigned char*)d_in[3];  // jnp.bool_ -> 1 byte
    const float*         W        = (const float*)d_in[4];
    const float*         bias     = (const float*)d_in[5];
    float*               out      = (float*)d_out;

    const int n_rows  = in_sizes[1];                 // N (half-edges)
    const int n_tiles = (n_rows + TILE_M - 1) / TILE_M;

    int blocks = (n_tiles + WAVES_PER_BLOCK - 1) / WAVES_PER_BLOCK;
    if (blocks > MAX_BLOCKS) blocks = MAX_BLOCKS;
    if (blocks < 1) blocks = 1;

    heconv_wmma_kernel<<<blocks, BLOCK, 0, stream>>>(
        x, next_idx, twin_idx, has_twin, W, bias, out, n_rows, n_tiles);
}